// Bert_BiLSTM_CRF_26268019982424
// MI455X (gfx1250) — compile-verified
//
#include <hip/hip_runtime.h>
#include <stdint.h>

// Viterbi decode (max-plus semiring) for B=256, T=4096, K=6.
// Strategy: chunked associative scan over max-plus transfer matrices.
//   Phase 1: 64 chunks/batch compute 6x6 chunk transfer matrices (lane k owns
//            column k -> no cross-lane deps), operands streamed via
//            global_load_async_to_lds_b128/b64 double buffering (ASYNCcnt).
//   Phase 2: 64-step boundary scan (tiny, one lane).
//   Phase 3: chunks re-walk locally, emitting backpointers to LDS (6-lane
//            row-parallel with wave32 shuffles for delta broadcast).
//   Phase 4: backtrack = map composition, chunk-parallel, then emit path.
// WMMA does not apply (tropical semiring: max, not accumulate). The CDNA5
// paths used here are the async-tensor copies + split wait counters.
// Bandwidth: 2x ~176MB streamed; transitions (151MB) fit the 192MB L2, so the
// phase-3 re-read mostly hits L2. Critical chains cut from 4096 to 64 steps.

#define TT 4096
#define KK 6
#define NC 64          // chunks per batch
#define CL 64          // steps per chunk (last chunk has CL-1: steps 1..4095)
#define W  2           // steps per LDS tile
#define NTILES (CL / W)
#define NTHREADS 512   // 16 waves, 4 groups of 8 lanes per wave, 64 groups
#define START_TAG 4

#define TILE_F (W * 36 + W * 6)   // 84 floats per chunk per buffer

__device__ __forceinline__ unsigned ldsOffset(const void* p) {
  // generic 'shared' pointer low 32 bits == LDS offset (ISA 10.2 aperture map)
  return (unsigned)(unsigned long long)p;
}

__device__ __forceinline__ void asyncLoadB128(unsigned lds, const float* g) {
  asm volatile("global_load_async_to_lds_b128 %0, %1, off"
               :: "v"(lds), "v"(g) : "memory");
}
__device__ __forceinline__ void asyncLoadB64(unsigned lds, const float* g) {
  asm volatile("global_load_async_to_lds_b64 %0, %1, off"
               :: "v"(lds), "v"(g) : "memory");
}
__device__ __forceinline__ void waitAsyncLE4() {
  asm volatile("s_wait_asynccnt 4" ::: "memory");
}
__device__ __forceinline__ void waitAsync0() {
  asm volatile("s_wait_asynccnt 0" ::: "memory");
}

__global__ __launch_bounds__(NTHREADS)
void viterbi_crf_kernel(const float* __restrict__ feats,
                        const float* __restrict__ trans,
                        float* __restrict__ out)
{
  __shared__ __align__(16) float tileBuf[NC][2][TILE_F]; // 43008 B
  __shared__ float Pmat[NC][36];                         //  9216 B
  __shared__ float Earr[NC][8];                          //  2048 B
  __shared__ unsigned char psiB[TT][8];                  // 32768 B
  __shared__ unsigned char Gmap[NC][8];                  //   512 B
  __shared__ int thArr[NC];                              //   256 B
  __shared__ int lastTagSh;

  const int b          = blockIdx.x;
  const int B          = gridDim.x;
  const int tid        = threadIdx.x;
  const int chunk      = tid >> 3;        // 0..63
  const int sub        = tid & 7;         // lane within group (0..7; 0..5 active)
  const int laneInWave = tid & 31;
  const int waveChunk0 = (tid >> 5) * 4;  // first chunk handled by this wave
  const size_t bT      = (size_t)b * TT;
  const int nsteps     = (chunk == NC - 1) ? (CL - 1) : CL;

  // ---- async tile loader: per wave, 4 groups x (18 b128 trans + 6 b64 feat)
  // Exactly 4 async instructions per wave per tile (3x b128 sweep + 1x b64),
  // so s_wait_asynccnt 4 == "previous tile fully landed". EXEC-masked async
  // VMEM still issues while ASYNCcnt != 0 (ISA 4.2.2), keeping counts uniform.
  auto issueTile = [&](int tileIdx, int buf) {
    #pragma unroll
    for (int i = 0; i < 3; ++i) {
      int c = i * 32 + laneInWave;
      if (c < 72) {                       // 4 groups * 18 b128 chunks
        int g = c / 18, q = c % 18;
        int ck = waveChunk0 + g;
        int s0 = ck * CL + 1 + tileIdx * W;
        int vs = TT - s0; if (vs > W) vs = W;   // valid steps in this tile
        if (q < vs * 9) {                 // 9 b128 chunks per step (144 B)
          const float* gp = trans + (bT + (size_t)s0) * 36 + q * 4;
          asyncLoadB128(ldsOffset(&tileBuf[ck][buf][q * 4]), gp);
        }
      }
    }
    {
      int c = laneInWave;
      if (c < 24) {                       // 4 groups * 6 b64 chunks
        int g = c / 6, q = c % 6;
        int ck = waveChunk0 + g;
        int s0 = ck * CL + 1 + tileIdx * W;
        int vs = TT - s0; if (vs > W) vs = W;
        if (q < vs * 3) {                 // 3 b64 chunks per step (24 B)
          const float* gp = feats + (bT + (size_t)s0) * 6 + q * 2;
          asyncLoadB64(ldsOffset(&tileBuf[ck][buf][72 + q * 2]), gp);
        }
      }
    }
  };

  // ================= Phase 1: chunk transfer-matrix products ==============
  // Lane k (sub) owns column k of running product P: P[j] = prod[j][k].
  // step: newP[j] = feat[j] + max_m( trans[j][m] + P[m] ).  No cross-lane deps.
  {
    float P[6];
    issueTile(0, 0);
    for (int t = 0; t < NTILES; ++t) {
      if (t + 1 < NTILES) { issueTile(t + 1, (t + 1) & 1); waitAsyncLE4(); }
      else                { waitAsync0(); }
      const float* tb = &tileBuf[chunk][t & 1][0];
      #pragma unroll
      for (int i = 0; i < W; ++i) {
        int sL = t * W + i;
        if (sL >= nsteps) break;
        const float* tr = tb + i * 36;       // trans[s][j][m], broadcast reads
        const float* ft = tb + 72 + i * 6;   // feat[s][j]
        if (sL == 0) {
          #pragma unroll
          for (int j = 0; j < 6; ++j) P[j] = tr[j * 6 + (sub < 6 ? sub : 0)] + ft[j];
        } else {
          float n[6];
          #pragma unroll
          for (int j = 0; j < 6; ++j) {
            float v = tr[j * 6 + 0] + P[0];
            #pragma unroll
            for (int m = 1; m < 6; ++m) v = fmaxf(v, tr[j * 6 + m] + P[m]);
            n[j] = v + ft[j];
          }
          #pragma unroll
          for (int j = 0; j < 6; ++j) P[j] = n[j];
        }
      }
    }
    if (sub < 6) {
      #pragma unroll
      for (int j = 0; j < 6; ++j) Pmat[chunk][j * 6 + sub] = P[j];
    }
  }
  __syncthreads();

  // ================= Phase 2: boundary scan over 64 chunk matrices ========
  if (tid == 0) {
    float e[6];
    #pragma unroll
    for (int j = 0; j < 6; ++j) e[j] = -10000.0f;
    e[START_TAG] = 0.0f;
    for (int c = 0; c < NC; ++c) {
      #pragma unroll
      for (int j = 0; j < 6; ++j) Earr[c][j] = e[j];   // delta at t = c*64
      float n[6];
      #pragma unroll
      for (int j = 0; j < 6; ++j) {
        float v = Pmat[c][j * 6 + 0] + e[0];
        #pragma unroll
        for (int m = 1; m < 6; ++m) v = fmaxf(v, Pmat[c][j * 6 + m] + e[m]);
        n[j] = v;
      }
      #pragma unroll
      for (int j = 0; j < 6; ++j) e[j] = n[j];
    }
    float best = e[0]; int bi = 0;
    #pragma unroll
    for (int j = 1; j < 6; ++j) if (e[j] > best) { best = e[j]; bi = j; }
    out[b] = best;            // score[b] = max_j delta_{T-1}[j]
    lastTagSh = bi;
  }
  __syncthreads();

  // ================= Phase 3: re-walk chunks, emit backpointers ===========
  // Lane j owns delta[j]; delta vector broadcast via wave32 shuffles.
  {
    const int gbase = laneInWave & 24;
    float dj = (sub < 6) ? Earr[chunk][sub] : -10000.0f;
    issueTile(0, 0);
    for (int t = 0; t < NTILES; ++t) {
      if (t + 1 < NTILES) { issueTile(t + 1, (t + 1) & 1); waitAsyncLE4(); }
      else                { waitAsync0(); }
      const float* tb = &tileBuf[chunk][t & 1][0];
      #pragma unroll
      for (int i = 0; i < W; ++i) {
        int sL = t * W + i;
        const float* tr = tb + i * 36;
        const float* ft = tb + 72 + i * 6;
        float best = -3.0e38f; int bidx = 0;
        #pragma unroll
        for (int k = 0; k < 6; ++k) {            // shuffles run uniformly
          float dk = __shfl(dj, gbase + k, 32);
          float s  = tr[(sub < 6 ? sub : 0) * 6 + k] + dk;
          if (s > best) { best = s; bidx = k; }  // '>' keeps first max (jnp)
        }
        if (sL < nsteps && sub < 6) {
          int sG = chunk * CL + 1 + sL;
          psiB[sG][sub] = (unsigned char)bidx;
          dj = best + ft[sub];
        }
      }
    }
  }
  __syncthreads();

  // ================= Phase 4a: compose backward maps per chunk ============
  if (sub < 6) {
    int sHi = (chunk == NC - 1) ? (TT - 1) : (chunk + 1) * CL;
    int sLo = chunk * CL + 1;
    int f = sub;
    for (int s = sHi; s >= sLo; --s) f = psiB[s][f];
    Gmap[chunk][sub] = (unsigned char)f;
  }
  __syncthreads();

  // ================= Phase 4b: chunk-boundary tag scan ====================
  if (tid == 0) {
    int tg = lastTagSh;
    thArr[NC - 1] = tg;                 // tag at s_hi of chunk c
    for (int c = NC - 1; c >= 1; --c) {
      tg = Gmap[c][tg];
      thArr[c - 1] = tg;
    }
  }
  __syncthreads();

  // ================= Phase 4c: emit path ==================================
  if (sub == 0) {
    float* path = out + B;
    int sHi = (chunk == NC - 1) ? (TT - 1) : (chunk + 1) * CL;
    int sLo = chunk * CL + 1;
    int tg = thArr[chunk];
    if (chunk == NC - 1) path[bT + (TT - 1)] = (float)tg;
    for (int s = sHi; s >= sLo; --s) {
      tg = psiB[s][tg];
      path[bT + (s - 1)] = (float)tg;
    }
  }
}

extern "C" void kernel_launch(void* const* d_in, const int* in_sizes, int n_in,
                              void* d_out, int out_size, void* d_ws, size_t ws_size,
                              hipStream_t stream) {
  const float* feats = (const float*)d_in[0];   // [B, T, K]
  const float* trans = (const float*)d_in[1];   // [B, T, K, K]
  float* out = (float*)d_out;                   // score[B] ++ path[B*T]
  (void)d_ws; (void)ws_size; (void)n_in; (void)out_size;
  int B = in_sizes[0] / (TT * KK);
  viterbi_crf_kernel<<<B, NTHREADS, 0, stream>>>(feats, trans, out);
}